// DinAttn_4243427688477
// MI455X (gfx1250) — compile-verified
//
#include <hip/hip_runtime.h>

#define L_DIM   200
#define D_DIM   128
#define H_DIM   256
#define M_TILES 13      // ceil(200/16)
#define BSTR    264     // B-matrix LDS row stride (pad 8 -> lane halves hit disjoint bank sets)
#define AXSTR   132     // x-tile LDS row stride (pad 4 -> conflict-free b64 fragment loads)

typedef __attribute__((ext_vector_type(2))) float v2f;
typedef __attribute__((ext_vector_type(8))) float v8f;

__global__ __launch_bounds__(256) void din_attn_kernel(
    const float* __restrict__ target,
    const float* __restrict__ item_seq,
    const int*   __restrict__ seq_len,
    const float* __restrict__ W1,
    const float* __restrict__ b1,
    const float* __restrict__ W2,
    const float* __restrict__ b2,
    float*       __restrict__ out)
{
    // ~290 KB of the WGP's 320 KB LDS.
    __shared__ __align__(16) float Bmat[H_DIM * BSTR];  // rows 0..127 = W1b+W1c, 128..255 = W1d
    __shared__ __align__(16) float Ax[2][16 * AXSTR];   // ping-pong x tiles (async-DMA filled)
    __shared__ __align__(16) float tv[D_DIM];           // target vector for this b
    __shared__ float tvec[H_DIM];                       // b1 + t @ (W1a - W1c)
    __shared__ float w2s[H_DIM];
    __shared__ float scores[16];
    __shared__ float outred[D_DIM];

    const int b    = blockIdx.x;
    const int tid  = threadIdx.x;
    const int lane = tid & 31;
    const int wave = tid >> 5;     // 0..7
    const int m16  = lane & 15;    // M index within fragment
    const int kh   = lane >> 4;    // K half (ISA 16x4 f32 A/B layout)

    const float* __restrict__ sbase = item_seq + (size_t)b * L_DIM * D_DIM;

    // Issue the async DMA for one 16-row x tile into Ax[buf] (ASYNCcnt-tracked).
    auto issue_tile_async = [&](int mt_ld, int buf) {
#pragma unroll
        for (int i = 0; i < 2; ++i) {
            int idx = tid + i * 256;
            int r = idx >> 5, c = (idx & 31) << 2;
            int l = mt_ld * 16 + r;
            int lc = l < L_DIM ? l : L_DIM - 1;           // clamp; garbage rows masked later
            unsigned ldsoff = (unsigned)(size_t)&Ax[buf][r * AXSTR + c];
            unsigned voff   = (unsigned)((lc * D_DIM + c) * 4);
            asm volatile("global_load_async_to_lds_b128 %0, %1, %2"
                         :: "v"(ldsoff), "v"(voff), "s"(sbase) : "memory");
        }
    };

    // Kick off tile 0 DMA immediately; it overlaps all the setup below.
    issue_tile_async(0, 0);

    if (tid < D_DIM) tv[tid] = target[(size_t)b * D_DIM + tid];
    w2s[tid] = W2[tid];
    __syncthreads();

    // Per-b constant term: tvec[h] = b1[h] + sum_k t[k] * (W1a[k][h] - W1c[k][h])
    {
        float acc = b1[tid];
#pragma unroll 4
        for (int k = 0; k < D_DIM; ++k)
            acc += tv[k] * (W1[(size_t)k * H_DIM + tid] -
                            W1[(size_t)(256 + k) * H_DIM + tid]);
        tvec[tid] = acc;
    }

    // Stage fused B matrix into LDS: [W1b + W1c ; W1d], coalesced float4 loads (L2-hot).
    for (int idx = tid; idx < H_DIM * 64; idx += 256) {
        int r = idx >> 6, c = (idx & 63) << 2;
        float4 v;
        if (r < 128) {
            float4 vb = *(const float4*)&W1[(size_t)(128 + r) * H_DIM + c];
            float4 vc = *(const float4*)&W1[(size_t)(256 + r) * H_DIM + c];
            v.x = vb.x + vc.x; v.y = vb.y + vc.y; v.z = vb.z + vc.z; v.w = vb.w + vc.w;
        } else {
            v = *(const float4*)&W1[(size_t)(384 + (r - 128)) * H_DIM + c];
        }
        *(float4*)&Bmat[r * BSTR + c] = v;
    }

    // Each wave owns two N-tiles of 16 columns.
    const int   n0  = wave * 32 + m16;
    const int   n1  = n0 + 16;
    const float b2s = b2[0];
    const int   len = seq_len[b];

    const int d  = tid & 127;   // output channel this thread reduces
    const int rh = tid >> 7;    // row half (rows 0-7 vs 8-15)
    float oacc = 0.f;

    __syncthreads();            // Bmat + tvec ready
    const float tv0 = tvec[n0], tv1 = tvec[n1];
    const float w20 = w2s[n0],  w21 = w2s[n1];

    const float* __restrict__ B0   = &Bmat[(2 * kh) * BSTR + n0];
    const float* __restrict__ B1   = &Bmat[(2 * kh) * BSTR + n1];
    const float* __restrict__ B0h  = B0 + 128 * BSTR;   // W1d half
    const float* __restrict__ B1h  = B1 + 128 * BSTR;
    const float* __restrict__ tvb  = &tv[2 * kh];

    for (int mt = 0; mt < M_TILES; ++mt) {
        const int cur = mt & 1;

        // Wait for this wave's DMA of Ax[cur], then make it visible block-wide.
        asm volatile("s_wait_asynccnt 0" ::: "memory");
        __syncthreads();

        if (tid < 16) scores[tid] = 0.f;
        // Prefetch next tile into the other buffer; lands while we run the GEMM.
        if (mt + 1 < M_TILES) issue_tile_async(mt + 1, cur ^ 1);

        const float* __restrict__ Axc = &Ax[cur][m16 * AXSTR + 2 * kh];

        v8f acc0 = {0.f, 0.f, 0.f, 0.f, 0.f, 0.f, 0.f, 0.f};
        v8f acc1 = {0.f, 0.f, 0.f, 0.f, 0.f, 0.f, 0.f, 0.f};

        // x @ (W1b + W1c): K = 128 in steps of 4
#pragma unroll 8
        for (int kt = 0; kt < 32; ++kt) {
            v2f a = *(const v2f*)(Axc + 4 * kt);          // A[m][k0+2h .. +1]
            const float* bp0 = B0 + (4 * kt) * BSTR;
            const float* bp1 = B1 + (4 * kt) * BSTR;
            v2f bf0, bf1;
            bf0.x = bp0[0]; bf0.y = bp0[BSTR];
            bf1.x = bp1[0]; bf1.y = bp1[BSTR];
            acc0 = __builtin_amdgcn_wmma_f32_16x16x4_f32(
                false, a, false, bf0, (short)0, acc0, false, false);
            acc1 = __builtin_amdgcn_wmma_f32_16x16x4_f32(
                false, a, false, bf1, (short)0, acc1, false, false);
        }
        // (x * t) @ W1d: scale the A fragment in-register (t-pair broadcast from LDS),
        // so x*t is never materialized.
#pragma unroll 8
        for (int kt = 0; kt < 32; ++kt) {
            v2f a  = *(const v2f*)(Axc + 4 * kt);
            v2f tp = *(const v2f*)(tvb + 4 * kt);
            a = a * tp;
            const float* bp0 = B0h + (4 * kt) * BSTR;
            const float* bp1 = B1h + (4 * kt) * BSTR;
            v2f bf0, bf1;
            bf0.x = bp0[0]; bf0.y = bp0[BSTR];
            bf1.x = bp1[0]; bf1.y = bp1[BSTR];
            acc0 = __builtin_amdgcn_wmma_f32_16x16x4_f32(
                false, a, false, bf0, (short)0, acc0, false, false);
            acc1 = __builtin_amdgcn_wmma_f32_16x16x4_f32(
                false, a, false, bf1, (short)0, acc1, false, false);
        }

        // Fused epilogue: score[m] += sum_n relu(C[m][n] + tvec[n]) * W2[n]
        // C layout: VGPR r, lanes 0-15 -> (M=r, N=m16), lanes 16-31 -> (M=r+8, N=m16).
        float vsum[8];
#pragma unroll
        for (int r = 0; r < 8; ++r) {
            float h0 = acc0[r] + tv0; h0 = h0 > 0.f ? h0 : 0.f;
            float h1 = acc1[r] + tv1; h1 = h1 > 0.f ? h1 : 0.f;
            vsum[r] = h0 * w20 + h1 * w21;
        }
        // Butterfly over the 16-lane N groups: stage-major so the 8 bpermutes per
        // stage issue back-to-back (one LDS round-trip per stage, not per value).
#pragma unroll
        for (int st = 1; st <= 8; st <<= 1) {
            float t[8];
#pragma unroll
            for (int r = 0; r < 8; ++r) t[r] = __shfl_xor(vsum[r], st, 32);
#pragma unroll
            for (int r = 0; r < 8; ++r) vsum[r] += t[r];
        }
        if (m16 == 0) {
#pragma unroll
            for (int r = 0; r < 8; ++r)
                atomicAdd(&scores[r + 8 * kh], vsum[r]);    // ds_add_f32
        }
        __syncthreads();

        // out[b] += sum_m mask(l) * (score[m] + b2) * x[m][:]  (x already in LDS)
#pragma unroll
        for (int r = 0; r < 8; ++r) {
            int m = rh * 8 + r;
            int l = mt * 16 + m;
            float s = (l < len) ? (scores[m] + b2s) : 0.f;
            oacc += s * Ax[cur][m * AXSTR + d];
        }
        __syncthreads();
    }

    // Combine the two row-half accumulators per channel and store.
    if (rh == 1) outred[d] = oacc;
    __syncthreads();
    if (rh == 0) out[(size_t)b * D_DIM + d] = oacc + outred[d];
}

extern "C" void kernel_launch(void* const* d_in, const int* in_sizes, int n_in,
                              void* d_out, int out_size, void* d_ws, size_t ws_size,
                              hipStream_t stream)
{
    const float* target   = (const float*)d_in[0];
    const float* item_seq = (const float*)d_in[1];
    const int*   seq_len  = (const int*)  d_in[2];
    const float* W1       = (const float*)d_in[3];
    const float* b1       = (const float*)d_in[4];
    const float* W2       = (const float*)d_in[5];
    const float* b2       = (const float*)d_in[6];
    float*       out      = (float*)d_out;

    const int B = in_sizes[2];  // seq_len has one entry per batch element
    din_attn_kernel<<<dim3(B), dim3(256), 0, stream>>>(
        target, item_seq, seq_len, W1, b1, W2, b2, out);
}